// TinyGAT_86002425135784
// MI455X (gfx1250) — compile-verified
//
#include <hip/hip_runtime.h>
#include <hip/hip_bf16.h>

typedef __attribute__((ext_vector_type(2))) float v2f;
typedef __attribute__((ext_vector_type(8))) float v8f;

// ---------------------------------------------------------------------------
// float atomic max via integer punning (works for negative values too)
// ---------------------------------------------------------------------------
__device__ inline void atomicMaxFloat(float* addr, float val) {
    int iv = __float_as_int(val);
    if (iv >= 0) {
        atomicMax((int*)addr, iv);
    } else {
        atomicMin((unsigned int*)addr, (unsigned int)iv);
    }
}

// ---------------------------------------------------------------------------
// Fill kernel (workspace init; hipMemsetAsync avoided for graph safety)
// ---------------------------------------------------------------------------
__global__ void fill_kernel(float* __restrict__ p, float val, size_t n) {
    size_t i = (size_t)blockIdx.x * blockDim.x + threadIdx.x;
    if (i < n) p[i] = val;
}

// ---------------------------------------------------------------------------
// WMMA f32 GEMM: C[M,Nc] = A[M,K] @ B[K,Nc] (+ bias). One wave per 16x64 tile.
// Requires M%16==0, K%4==0, Nc%64==0 (true for all launches here).
// Uses V_WMMA_F32_16X16X4_F32; exact f32 precision.
// ---------------------------------------------------------------------------
__global__ void wmma_gemm_f32(const float* __restrict__ A, const float* __restrict__ B,
                              const float* __restrict__ bias, float* __restrict__ C,
                              int M, int K, int Nc) {
    int wid  = (int)((blockIdx.x * (size_t)blockDim.x + threadIdx.x) >> 5);
    int lane = threadIdx.x & 31;
    int mtiles  = M >> 4;
    int ntiles  = Nc >> 6;
    if (wid >= mtiles * ntiles) return;          // wave-uniform: EXEC stays all-1s
    int mt = wid % mtiles;
    int nt = wid / mtiles;
    int row0 = mt << 4;
    int col0 = nt << 6;

    int m16   = lane & 15;          // A row within tile / B,C col within tile
    int khalf = (lane >> 4) << 1;   // 0 or 2: which K pair this half-wave holds

    v8f acc0 = {}, acc1 = {}, acc2 = {}, acc3 = {};
    const float* Arow = A + (size_t)(row0 + m16) * K + khalf;

    for (int k0 = 0; k0 < K; k0 += 4) {
        v2f a;
        a.x = Arow[k0 + 0];
        a.y = Arow[k0 + 1];
        const float* Bp = B + (size_t)(k0 + khalf) * Nc + col0 + m16;
        v2f b0, b1, b2, b3;
        b0.x = Bp[0];   b0.y = Bp[Nc];
        b1.x = Bp[16];  b1.y = Bp[Nc + 16];
        b2.x = Bp[32];  b2.y = Bp[Nc + 32];
        b3.x = Bp[48];  b3.y = Bp[Nc + 48];
        acc0 = __builtin_amdgcn_wmma_f32_16x16x4_f32(false, a, false, b0, (short)0, acc0, false, false);
        acc1 = __builtin_amdgcn_wmma_f32_16x16x4_f32(false, a, false, b1, (short)0, acc1, false, false);
        acc2 = __builtin_amdgcn_wmma_f32_16x16x4_f32(false, a, false, b2, (short)0, acc2, false, false);
        acc3 = __builtin_amdgcn_wmma_f32_16x16x4_f32(false, a, false, b3, (short)0, acc3, false, false);
    }

    // C/D layout: VGPR v -> row v + 8*(lane>>4), col = lane&15 (+16 per N tile)
    int rbase = row0 + ((lane >> 4) << 3);
    int cb    = col0 + m16;
    float bb0 = bias ? bias[cb]      : 0.f;
    float bb1 = bias ? bias[cb + 16] : 0.f;
    float bb2 = bias ? bias[cb + 32] : 0.f;
    float bb3 = bias ? bias[cb + 48] : 0.f;
#pragma unroll
    for (int v = 0; v < 8; ++v) {
        size_t ro = (size_t)(rbase + v) * Nc + cb;
        C[ro]      = acc0[v] + bb0;
        C[ro + 16] = acc1[v] + bb1;
        C[ro + 32] = acc2[v] + bb2;
        C[ro + 48] = acc3[v] + bb3;
    }
}

// ---------------------------------------------------------------------------
// GATv2 per-edge score: e[h] = sum_c leakyrelu(xl[src][h][c]+xr[dst][h][c])*att[h][c]
// One wave per edge; lane handles channels 2*lane, 2*lane+1 (C=64).
// Also atomic-max into per-dst running max (segment_max).
// ---------------------------------------------------------------------------
template <int H>
__global__ void gat_score_kernel(const float* __restrict__ xl, const float* __restrict__ xr,
                                 const float* __restrict__ att,
                                 const long long* __restrict__ ei, int E, int ET,
                                 float* __restrict__ escore, float* __restrict__ mmax) {
    int wid  = (int)((blockIdx.x * (size_t)blockDim.x + threadIdx.x) >> 5);
    int lane = threadIdx.x & 31;
    if (wid >= ET) return;
    long long s, d;
    if (wid < E) { s = ei[wid]; d = ei[E + wid]; } else { s = d = wid - E; }
    const float* xls = xl + (size_t)s * (H * 64);
    const float* xrd = xr + (size_t)d * (H * 64);
#pragma unroll
    for (int h = 0; h < H; ++h) {
        int off = h * 64 + lane * 2;
        float2 a = *(const float2*)(xls + off);
        float2 b = *(const float2*)(xrd + off);
        float2 w = *(const float2*)(att + off);
        float u = a.x + b.x; u = u > 0.f ? u : 0.2f * u;
        float v = a.y + b.y; v = v > 0.f ? v : 0.2f * v;
        float sc = u * w.x + v * w.y;
#pragma unroll
        for (int m = 16; m >= 1; m >>= 1) sc += __shfl_xor(sc, m, 32);
        if (lane == 0) {
            escore[(size_t)wid * H + h] = sc;
            atomicMaxFloat(&mmax[(size_t)d * H + h], sc);
        }
    }
}

// ---------------------------------------------------------------------------
// ex = exp(e - m[dst]); denom[dst] += ex   (segment softmax numerator/denominator)
// One thread per (edge, head).
// ---------------------------------------------------------------------------
template <int H>
__global__ void gat_exp_kernel(const long long* __restrict__ ei, int E, int ET,
                               float* __restrict__ escore, const float* __restrict__ mmax,
                               float* __restrict__ denom) {
    int idx = (int)(blockIdx.x * (size_t)blockDim.x + threadIdx.x);
    if (idx >= ET * H) return;
    int e = idx / H;
    int h = idx - e * H;
    long long d = (e < E) ? ei[E + e] : (long long)(e - E);
    float ex = __expf(escore[idx] - mmax[(size_t)d * H + h]);
    escore[idx] = ex;
    atomicAdd(&denom[(size_t)d * H + h], ex);
}

// ---------------------------------------------------------------------------
// out[dst] += xl[src] * (ex / (denom[dst]+1e-16)); one wave per edge.
// ---------------------------------------------------------------------------
template <int H>
__global__ void gat_agg_kernel(const float* __restrict__ xl, const float* __restrict__ escore,
                               const float* __restrict__ denom,
                               const long long* __restrict__ ei, int E, int ET,
                               float* __restrict__ out) {
    int wid  = (int)((blockIdx.x * (size_t)blockDim.x + threadIdx.x) >> 5);
    int lane = threadIdx.x & 31;
    if (wid >= ET) return;
    long long s, d;
    if (wid < E) { s = ei[wid]; d = ei[E + wid]; } else { s = d = wid - E; }
    const float* xls = xl + (size_t)s * (H * 64);
    float*       od  = out + (size_t)d * (H * 64);
#pragma unroll
    for (int h = 0; h < H; ++h) {
        float alpha = escore[(size_t)wid * H + h] /
                      (denom[(size_t)d * H + h] + 1e-16f);
        int off = h * 64 + lane * 2;
        float2 a = *(const float2*)(xls + off);
        atomicAdd(&od[off],     a.x * alpha);
        atomicAdd(&od[off + 1], a.y * alpha);
    }
}

// ---------------------------------------------------------------------------
// h = elu(h + bias)
// ---------------------------------------------------------------------------
__global__ void bias_elu_kernel(float* __restrict__ h, const float* __restrict__ bias,
                                int cols, size_t n) {
    size_t i = (size_t)blockIdx.x * blockDim.x + threadIdx.x;
    if (i >= n) return;
    float v = h[i] + bias[i % (size_t)cols];
    h[i] = v > 0.f ? v : (__expf(v) - 1.f);
}

// ---------------------------------------------------------------------------
// Launch pipeline
// ---------------------------------------------------------------------------
extern "C" void kernel_launch(void* const* d_in, const int* in_sizes, int n_in,
                              void* d_out, int out_size, void* d_ws, size_t ws_size,
                              hipStream_t stream) {
    const float*      x    = (const float*)d_in[0];
    const long long*  ei   = (const long long*)d_in[1];
    const float*      W1l  = (const float*)d_in[2];
    const float*      W1r  = (const float*)d_in[3];
    const float*      att1 = (const float*)d_in[4];
    const float*      b1   = (const float*)d_in[5];
    const float*      W2l  = (const float*)d_in[6];
    const float*      W2r  = (const float*)d_in[7];
    const float*      att2 = (const float*)d_in[8];
    const float*      b2   = (const float*)d_in[9];
    const float*      Wlin = (const float*)d_in[10];
    const float*      blin = (const float*)d_in[11];
    float*            out  = (float*)d_out;

    const int N  = in_sizes[0] / 128;  // 50000
    const int E  = in_sizes[1] / 2;    // 800000
    const int ET = E + N;              // edges + self loops

    auto cdiv = [](long long a, long long b) { return (int)((a + b - 1) / b); };
    const int BS = 256;

    // workspace layout (floats)
    float* ws = (float*)d_ws;
    size_t o = 0;
    float* xl1 = ws + o; o += (size_t)N * 256;
    float* xr1 = ws + o; o += (size_t)N * 256;
    float* h1  = ws + o; o += (size_t)N * 256;   // agg -> elu in place
    float* e1  = ws + o; o += (size_t)ET * 4;    // score -> ex in place
    float* m1  = ws + o; o += (size_t)N * 4;
    float* d1  = ws + o; o += (size_t)N * 4;
    // layer-2 buffers alias xr1 (xr1 is dead after layer-1 scoring)
    float* xl2 = xr1;
    float* xr2 = xl2 + (size_t)N * 64;
    float* h2  = xr2 + (size_t)N * 64;
    float* e2  = h2  + (size_t)N * 64;
    float* m2  = e2  + (size_t)ET;
    float* d2  = m2  + (size_t)N;

    const float NEG_INF = -__builtin_huge_valf();

    // ---- Layer 1 ----
    fill_kernel<<<cdiv((size_t)N * 256, BS), BS, 0, stream>>>(h1, 0.f, (size_t)N * 256);
    fill_kernel<<<cdiv((size_t)N * 4, BS), BS, 0, stream>>>(m1, NEG_INF, (size_t)N * 4);
    fill_kernel<<<cdiv((size_t)N * 4, BS), BS, 0, stream>>>(d1, 0.f, (size_t)N * 4);

    {   // xl1 = x @ W1l ; xr1 = x @ W1r   (M=N, K=128, Nc=256)
        int waves = (N / 16) * (256 / 64);
        int grid  = cdiv((long long)waves * 32, BS);
        wmma_gemm_f32<<<grid, BS, 0, stream>>>(x, W1l, nullptr, xl1, N, 128, 256);
        wmma_gemm_f32<<<grid, BS, 0, stream>>>(x, W1r, nullptr, xr1, N, 128, 256);
    }
    {
        int gridE = cdiv((long long)ET * 32, BS);
        gat_score_kernel<4><<<gridE, BS, 0, stream>>>(xl1, xr1, att1, ei, E, ET, e1, m1);
        gat_exp_kernel<4><<<cdiv((long long)ET * 4, BS), BS, 0, stream>>>(ei, E, ET, e1, m1, d1);
        gat_agg_kernel<4><<<gridE, BS, 0, stream>>>(xl1, e1, d1, ei, E, ET, h1);
    }
    bias_elu_kernel<<<cdiv((size_t)N * 256, BS), BS, 0, stream>>>(h1, b1, 256, (size_t)N * 256);

    // ---- Layer 2 ----
    fill_kernel<<<cdiv((size_t)N * 64, BS), BS, 0, stream>>>(h2, 0.f, (size_t)N * 64);
    fill_kernel<<<cdiv((size_t)N, BS), BS, 0, stream>>>(m2, NEG_INF, (size_t)N);
    fill_kernel<<<cdiv((size_t)N, BS), BS, 0, stream>>>(d2, 0.f, (size_t)N);

    {   // xl2 = h1 @ W2l ; xr2 = h1 @ W2r  (M=N, K=256, Nc=64)
        int waves = (N / 16) * (64 / 64);
        int grid  = cdiv((long long)waves * 32, BS);
        wmma_gemm_f32<<<grid, BS, 0, stream>>>(h1, W2l, nullptr, xl2, N, 256, 64);
        wmma_gemm_f32<<<grid, BS, 0, stream>>>(h1, W2r, nullptr, xr2, N, 256, 64);
    }
    {
        int gridE = cdiv((long long)ET * 32, BS);
        gat_score_kernel<1><<<gridE, BS, 0, stream>>>(xl2, xr2, att2, ei, E, ET, e2, m2);
        gat_exp_kernel<1><<<cdiv((long long)ET, BS), BS, 0, stream>>>(ei, E, ET, e2, m2, d2);
        gat_agg_kernel<1><<<gridE, BS, 0, stream>>>(xl2, e2, d2, ei, E, ET, h2);
    }
    bias_elu_kernel<<<cdiv((size_t)N * 64, BS), BS, 0, stream>>>(h2, b2, 64, (size_t)N * 64);

    // ---- Final linear: out = h2 @ Wlin + blin  (M=N, K=64, Nc=64) ----
    {
        int waves = (N / 16) * (64 / 64);
        int grid  = cdiv((long long)waves * 32, BS);
        wmma_gemm_f32<<<grid, BS, 0, stream>>>(h2, Wlin, blin, out, N, 64, 64);
    }
}